// BOXLoss_10187662426677
// MI455X (gfx1250) — compile-verified
//
#include <hip/hip_runtime.h>
#include <stdint.h>

#define CLAMPV 1e-4f
#define GRID_P 1024
#define BLOCK_P 256
#define WAVES_PER_BLOCK (BLOCK_P / 32)
#define TOTAL_WAVES (GRID_P * WAVES_PER_BLOCK)
#define NPART GRID_P   // one partial per block of pass 3

typedef __attribute__((ext_vector_type(2))) float v2f;
typedef __attribute__((ext_vector_type(8))) float v8f;

// s >= 0 always (scores are uniform [0,1)).  Matches jnp.power semantics
// incl. 0^0 = 1, 0^e = 0 for e > 0.
__device__ __forceinline__ float pow_nn(float s, float e) {
    return (s > 0.f) ? __expf(e * __logf(s)) : (e > 0.f ? 0.f : 1.f);
}

__global__ void k_init(unsigned* p) {           // launched with 64 threads
    p[threadIdx.x] = 0u;                        // 0u == bits of 0.0f
}

// Pass 1: conf[g] = max over masked rows of scores[n]  (>= 0, init 0 matches ref)
__global__ __launch_bounds__(BLOCK_P)
void k_pass1(const int* __restrict__ boxes, const float* __restrict__ scores,
             unsigned* __restrict__ conf_bits, int N) {
    __shared__ unsigned smax[32];
    const int lane = threadIdx.x & 31;
    const int wid  = threadIdx.x >> 5;
    if (threadIdx.x < 32) smax[threadIdx.x] = 0u;
    __syncthreads();
    float cmax = 0.f;
    for (int row = blockIdx.x * WAVES_PER_BLOCK + wid; row < N; row += TOTAL_WAVES) {
        int   v = boxes[(size_t)row * 32 + lane];   // coalesced 128B per row
        float s = scores[row];                      // broadcast load
        if (v > 0) cmax = fmaxf(cmax, s);
    }
    atomicMax(&smax[lane], __float_as_uint(cmax));  // non-neg float: bits order == value order
    __syncthreads();
    if (threadIdx.x < 32) atomicMax(&conf_bits[threadIdx.x], smax[threadIdx.x]);
}

// Pass 2: mx[g] = max over masked rows of raw = s^conf[g] * iou  (>= 0, init 0 ok)
__global__ __launch_bounds__(BLOCK_P)
void k_pass2(const int* __restrict__ boxes, const float* __restrict__ scores,
             const float* __restrict__ iou, const unsigned* __restrict__ conf_bits,
             unsigned* __restrict__ mx_bits, int N) {
    __shared__ unsigned smax[32];
    const int lane = threadIdx.x & 31;
    const int wid  = threadIdx.x >> 5;
    if (threadIdx.x < 32) smax[threadIdx.x] = 0u;
    __syncthreads();
    const float conf = __uint_as_float(conf_bits[lane]);  // lane g holds conf[g]
    float rmax = 0.f;
    for (int row = blockIdx.x * WAVES_PER_BLOCK + wid; row < N; row += TOTAL_WAVES) {
        int v = boxes[(size_t)row * 32 + lane];
        if (v > 0) {
            float s = scores[row];
            float q = iou[row];
            rmax = fmaxf(rmax, pow_nn(s, conf) * q);
        }
    }
    atomicMax(&smax[lane], __float_as_uint(rmax));
    __syncthreads();
    if (threadIdx.x < 32) atomicMax(&mx_bits[threadIdx.x], smax[threadIdx.x]);
}

// Pass 3: per-element losses, block-sum -> partials[block]
__global__ __launch_bounds__(BLOCK_P)
void k_pass3(const int* __restrict__ boxes, const float* __restrict__ scores,
             const float* __restrict__ iou, const float* __restrict__ logits,
             const unsigned* __restrict__ conf_bits, const unsigned* __restrict__ mx_bits,
             float* __restrict__ partials, int N) {
    __shared__ float sred[BLOCK_P];
    const int lane = threadIdx.x & 31;
    const int wid  = threadIdx.x >> 5;
    const float conf = __uint_as_float(conf_bits[lane]);
    const float mxg  = __uint_as_float(mx_bits[lane]);
    float acc = 0.f;
    for (int row = blockIdx.x * WAVES_PER_BLOCK + wid; row < N; row += TOTAL_WAVES) {
        int      v     = boxes[(size_t)row * 32 + lane];
        bool     m     = v > 0;
        unsigned rmask = __builtin_amdgcn_ballot_w32(m);   // full row mask (wave32, lane==g)
        float x = logits[row];
        float p = 1.f / (1.f + __expf(-x));
        p = fminf(fmaxf(p, CLAMPV), 1.f - CLAMPV);
        if (rmask == 0u) {
            // neg row: -log(1-p) * p^2 * (1 - FOCAL_ALPHA); count once per row
            if (lane == 0) acc += -__logf(1.f - p) * p * p * 0.75f;
        } else if (m) {
            float s   = scores[row];
            float q   = iou[row];
            float raw = pow_nn(s, conf) * q;
            float w   = (raw + CLAMPV) / (mxg + CLAMPV);   // > 0 whenever masked
            float wc  = fminf(fmaxf(w, CLAMPV), 1.f - CLAMPV);
            float a   = wc * (1.f - p);
            float b   = (1.f - wc) * p;
            acc += (-__logf(p) * a * a - __logf(1.f - p) * b * b) * 0.25f;
        }
    }
    sred[threadIdx.x] = acc;
    __syncthreads();
    for (int off = BLOCK_P / 2; off > 0; off >>= 1) {
        if (threadIdx.x < off) sred[threadIdx.x] += sred[threadIdx.x + off];
        __syncthreads();
    }
    if (threadIdx.x == 0) partials[blockIdx.x] = sred[0];
}

// Final: sum NPART partials with V_WMMA_F32_16X16X4_F32 (C += A x ones),
// divide by num_pos_avg.  One wave, no divergence before the WMMAs (EXEC all 1s).
__global__ void k_final(const float* __restrict__ partials, const int* __restrict__ npa,
                        float* __restrict__ out) {
    const int lane = threadIdx.x;          // 32 threads
    v8f c = {};
    v2f ones;
    ones.x = 1.f; ones.y = 1.f;
    #pragma unroll 1
    for (int chunk = 0; chunk < NPART / 64; ++chunk) {
        v2f a = *(const v2f*)(partials + chunk * 64 + lane * 2);
        // D = A(16x4) * ones(4x16) + C  ->  D[m,n] = sum_k A[m,k]  (same for all n)
        c = __builtin_amdgcn_wmma_f32_16x16x4_f32(false, a, false, ones,
                                                  (short)0, c, false, false);
    }
    float lanesum = c[0] + c[1] + c[2] + c[3] + c[4] + c[5] + c[6] + c[7];
    // lane 0 holds D[0..7,0], lane 16 holds D[8..15,0]
    float other = __shfl_xor(lanesum, 16, 32);
    if (lane == 0) out[0] = (lanesum + other) / (float)npa[0];
}

extern "C" void kernel_launch(void* const* d_in, const int* in_sizes, int n_in,
                              void* d_out, int out_size, void* d_ws, size_t ws_size,
                              hipStream_t stream) {
    const float* logits = (const float*)d_in[0];
    const float* scores = (const float*)d_in[1];
    const float* iou    = (const float*)d_in[2];
    const int*   boxes  = (const int*)d_in[3];
    const int*   npa    = (const int*)d_in[4];
    const int N = in_sizes[0];             // rows; G == 32 == wave width

    // scratch layout (4352 bytes total): conf[32] | mx[32] | partials[NPART]
    unsigned* conf_bits = (unsigned*)d_ws;
    unsigned* mx_bits   = conf_bits + 32;
    float*    partials  = (float*)(conf_bits + 64);

    k_init <<<1, 64, 0, stream>>>(conf_bits);   // zero conf+mx every call
    k_pass1<<<GRID_P, BLOCK_P, 0, stream>>>(boxes, scores, conf_bits, N);
    k_pass2<<<GRID_P, BLOCK_P, 0, stream>>>(boxes, scores, iou, conf_bits, mx_bits, N);
    k_pass3<<<GRID_P, BLOCK_P, 0, stream>>>(boxes, scores, iou, logits,
                                            conf_bits, mx_bits, partials, N);
    k_final<<<1, 32, 0, stream>>>(partials, npa, (float*)d_out);
}